// SpikingIPN_56057913147866
// MI455X (gfx1250) — compile-verified
//
#include <hip/hip_runtime.h>

// ---------------------------------------------------------------------------
// Izhikevich spiking populations (vIPN + dIPN), 4.19M neurons each, T=20.
// Memory-bound streaming: 671 MB of noise read exactly once -> ~29 us floor
// at 23.3 TB/s HBM. Noise loads are non-temporal (TH=NT) since the stream is
// 3.5x the 192 MB L2 and has zero reuse. Neuron state lives in registers
// (compiler packs the math into v_pk_fma_f32). WMMA f32 16x16x4 performs the
// deterministic 32-lane reduction of the firing-rate trace.
// ---------------------------------------------------------------------------

#define NPOP            4194304
#define T_STEPS         20
#define THREADS         256
#define PER_THREAD      4
#define NEURONS_PER_BLK (THREADS * PER_THREAD)        // 1024
#define BLOCKS_PER_POP  (NPOP / NEURONS_PER_BLK)      // 4096
#define ROW4            (NPOP / 4)                    // float4 per time row

typedef __attribute__((ext_vector_type(2))) float v2f;
typedef __attribute__((ext_vector_type(4))) float v4f;
typedef __attribute__((ext_vector_type(8))) float v8f;

// Deterministic 32-lane sum via two V_WMMA_F32_16X16X4_F32.
// WMMA #1: A[lane] = {x, 0}, B = ones  ->  D[m][n] = rowsum_m (x_m + x_{m+16}).
//   Lanes 0-15 hold rows 0-7 in d0..d7, lanes 16-31 hold rows 8-15, so the
//   per-lane sum s covers half the rows; s_low + s_high = total.
// WMMA #2: A[lane] = {s, s}, B = ones  ->  every D element = 2*(s_low+s_high).
__device__ __forceinline__ float wave_reduce_wmma(float x) {
  v2f a;    a.x = x;   a.y = 0.0f;
  v2f ones; ones.x = 1.0f; ones.y = 1.0f;
  v8f c = {};
  v8f d = __builtin_amdgcn_wmma_f32_16x16x4_f32(false, a, false, ones,
                                                (short)0, c, false, false);
  float s = ((d[0] + d[1]) + (d[2] + d[3])) + ((d[4] + d[5]) + (d[6] + d[7]));
  v2f a2; a2.x = s; a2.y = s;
  v8f d2 = __builtin_amdgcn_wmma_f32_16x16x4_f32(false, a2, false, ones,
                                                 (short)0, c, false, false);
  return d2[0] * 0.5f;
}

__global__ __launch_bounds__(THREADS)
void izh_sim_kernel(const float* __restrict__ mhb_p,
                    const float* __restrict__ lhb_p,
                    const float* __restrict__ av_p,
                    const float* __restrict__ noise_v,
                    const float* __restrict__ noise_d,
                    float* __restrict__ block_sums) {
  const int  b    = blockIdx.x;
  const bool is_d = (b >= BLOCKS_PER_POP);
  const int  pb   = is_d ? (b - BLOCKS_PER_POP) : b;

  const float mhb = mhb_p[0];
  const float lhb = lhb_p[0];
  const float av  = av_p[0];
  // I_drive + I_tonic, folded with the +140 constant of the v-update.
  const float idrive = is_d ? (lhb * 12.0f + av * 5.0f - 1.5f)
                            : (mhb * 20.0f + av * 10.0f - 1.0f);
  const float i140 = idrive + 140.0f;

  const v4f* __restrict__ noise4 = (const v4f*)(is_d ? noise_d : noise_v);
  const size_t gid = (size_t)pb * THREADS + threadIdx.x;  // float4 column

  // Issue all T loads up front: 20-deep MLP per thread, fully coalesced,
  // non-temporal (read-once stream, no L2 retention wanted).
  float nz[T_STEPS][4];
#pragma unroll
  for (int t = 0; t < T_STEPS; ++t) {
    v4f tmp = __builtin_nontemporal_load(&noise4[(size_t)t * ROW4 + gid]);
    nz[t][0] = tmp.x; nz[t][1] = tmp.y; nz[t][2] = tmp.z; nz[t][3] = tmp.w;
  }

  float v[PER_THREAD], u[PER_THREAD], r[PER_THREAD];
#pragma unroll
  for (int j = 0; j < PER_THREAD; ++j) {
    v[j] = -65.0f;          // V_INIT
    u[j] = -13.0f;          // B * V_INIT
    r[j] = 0.0f;
  }

#pragma unroll
  for (int t = 0; t < T_STEPS; ++t) {
#pragma unroll
    for (int j = 0; j < PER_THREAD; ++j) {
      const float vv   = v[j];
      const float base = (i140 + nz[t][j]) - u[j];
      // v1 = v + 0.04 v^2 + 5v + 140 - u + I  ->  0.04 v^2 + 6v + base
      const float v1 = (0.04f * vv) * vv + 6.0f * vv + base;
      // u1 = u + 0.02*(0.2 v - u) = 0.98 u + 0.004 v
      const float u1 = 0.98f * u[j] + 0.004f * vv;
      const bool  spiked = (v1 >= 30.0f);
      v[j] = spiked ? -65.0f : v1;                 // reset to C
      u[j] = u1 + (spiked ? 8.0f : 0.0f);          // + spk * D
      r[j] = r[j] * 0.9f + (spiked ? 0.1f : 0.0f); // EMA rate trace
    }
  }

  const float x = (r[0] + r[1]) + (r[2] + r[3]);

  // EXEC is all-ones here (no divergence above, exact grid) -> WMMA legal.
  const float wave_total = wave_reduce_wmma(x);

  __shared__ float smem[THREADS / 32];
  const int lane = threadIdx.x & 31;
  const int wv   = threadIdx.x >> 5;
  if (lane == 0) smem[wv] = wave_total;
  __syncthreads();
  if (threadIdx.x == 0) {
    float s = 0.0f;
#pragma unroll
    for (int w = 0; w < THREADS / 32; ++w) s += smem[w];
    block_sums[b] = s;
  }
}

__global__ __launch_bounds__(256)
void finalize_kernel(const float* __restrict__ block_sums,
                     const float* __restrict__ mhb_p,
                     const float* __restrict__ av_p,
                     const float* __restrict__ avmem_p,
                     float* __restrict__ out) {
  __shared__ float sv_s[256];
  __shared__ float sd_s[256];
  float sv = 0.0f, sd = 0.0f;
  for (int i = threadIdx.x; i < BLOCKS_PER_POP; i += 256) {
    sv += block_sums[i];
    sd += block_sums[BLOCKS_PER_POP + i];
  }
  sv_s[threadIdx.x] = sv;
  sd_s[threadIdx.x] = sd;
  __syncthreads();
  for (int off = 128; off > 0; off >>= 1) {
    if ((int)threadIdx.x < off) {
      sv_s[threadIdx.x] += sv_s[threadIdx.x + off];
      sd_s[threadIdx.x] += sd_s[threadIdx.x + off];
    }
    __syncthreads();
  }
  if (threadIdx.x == 0) {
    const float inv_n     = 1.0f / (float)NPOP;
    const float vipn_mean = sv_s[0] * inv_n;
    const float dipn_mean = sd_s[0] * inv_n;

    const float bi    = fminf(0.3f, vipn_mean * 3.0f);
    const float speed = 1.0f - bi;
    const float da    = -fminf(0.3f, dipn_mean * 2.0f);
    const float raphe = fminf(0.5f, (vipn_mean + dipn_mean) * 2.0f);
    const float avin  = fmaxf(mhb_p[0], av_p[0]) * 0.1f;
    const float avmem = fminf(1.0f, fmaxf(0.0f, avmem_p[0] * 0.95f + avin));

    out[0] = bi;
    out[1] = da;
    out[2] = raphe;
    out[3] = avmem;
    out[4] = vipn_mean;
    out[5] = dipn_mean;
    out[6] = speed;
  }
}

extern "C" void kernel_launch(void* const* d_in, const int* in_sizes, int n_in,
                              void* d_out, int out_size, void* d_ws,
                              size_t ws_size, hipStream_t stream) {
  (void)in_sizes; (void)n_in; (void)out_size; (void)ws_size;
  const float* mhb     = (const float*)d_in[0];
  const float* lhb     = (const float*)d_in[1];
  const float* av      = (const float*)d_in[2];
  const float* avmem   = (const float*)d_in[3];
  const float* noise_v = (const float*)d_in[4];
  const float* noise_d = (const float*)d_in[5];
  float* out        = (float*)d_out;
  float* block_sums = (float*)d_ws;  // 2 * 4096 floats = 32 KB

  izh_sim_kernel<<<2 * BLOCKS_PER_POP, THREADS, 0, stream>>>(
      mhb, lhb, av, noise_v, noise_d, block_sums);
  finalize_kernel<<<1, 256, 0, stream>>>(block_sums, mhb, av, avmem, out);
}